// GNNet_66614942761227
// MI455X (gfx1250) — compile-verified
//
#include <hip/hip_runtime.h>
#include <hip/hip_bf16.h>
#include <math.h>

// ---------------- problem constants (from reference) ----------------
#define N_NODES 1200
#define CS      7
#define EMB     64
#define OBS     9
#define NOBS    16
#define NE      38400
#define LOOPN   3
#define INV_TEMP 0.125f          // 1/sqrt(64)
#define LN_EPS   1e-6f

typedef float v2f __attribute__((ext_vector_type(2)));
typedef float v8f __attribute__((ext_vector_type(8)));

// --------------------------------------------------------------------
// Wave-level fp32 GEMM core: out[16 x 16*colTiles] = A[16 x K] * B[K x ...]
// V_WMMA_F32_16X16X4_F32, wave32, K multiple of 4, B must have all K rows.
// Lane layout (32-bit operands):
//   A frag: v0 = A[l&15][k+2*hf], v1 = A[l&15][k+1+2*hf]   (hf = lane>>4)
//   B frag: v0 = B[k+2*hf][n],    v1 = B[k+1+2*hf][n]
//   C frag: reg r, lane l -> row r+8*hf, col l&15
// k-outer / col-tile-inner with up to 4 accumulators so A fragments are
// loaded once per k-step.
// --------------------------------------------------------------------
__device__ __forceinline__ void wave_gemm16(
    const float* __restrict__ A, int lda,
    const float* __restrict__ B, int ldb, bool transB,
    int K, const float* __restrict__ bias, bool relu,
    float* __restrict__ out, int ldo, int colBase, int colTiles, int lane)
{
  const int hf  = (lane >> 4) & 1;
  const int l15 = lane & 15;
  v8f acc0 = {}, acc1 = {}, acc2 = {}, acc3 = {};
  for (int k = 0; k < K; k += 4) {
    const int ka = k + 2 * hf;
    v2f a;
    a.x = A[l15 * lda + ka];
    a.y = A[l15 * lda + ka + 1];
#define WG_STEP(NT, ACC)                                                     \
    {                                                                        \
      const int bcol = (NT) * 16 + l15;                                      \
      v2f b;                                                                 \
      if (transB) {                                                          \
        b.x = B[bcol * ldb + ka];                                            \
        b.y = B[bcol * ldb + ka + 1];                                        \
      } else {                                                               \
        b.x = B[(ka    ) * ldb + bcol];                                      \
        b.y = B[(ka + 1) * ldb + bcol];                                      \
      }                                                                      \
      ACC = __builtin_amdgcn_wmma_f32_16x16x4_f32(                           \
          false, a, false, b, (short)0, ACC, false, false);                  \
    }
    WG_STEP(0, acc0);
    if (colTiles > 1) WG_STEP(1, acc1);
    if (colTiles > 2) WG_STEP(2, acc2);
    if (colTiles > 3) WG_STEP(3, acc3);
#undef WG_STEP
  }
#define WG_OUT(NT, ACC)                                                      \
  if (colTiles > (NT)) {                                                     \
    const int bcol = (NT) * 16 + l15;                                        \
    for (int r = 0; r < 8; ++r) {                                            \
      int row = r + 8 * hf;                                                  \
      float res = ACC[r];                                                    \
      if (bias) res += bias[bcol];                                           \
      if (relu) res = fmaxf(res, 0.0f);                                      \
      out[row * ldo + colBase + bcol] = res;                                 \
    }                                                                        \
  }
  WG_OUT(0, acc0);
  WG_OUT(1, acc1);
  WG_OUT(2, acc2);
  WG_OUT(3, acc3);
#undef WG_OUT
}

// deterministic float atomic-max via sign-ordered integer atomics
__device__ __forceinline__ void atomicMaxF(float* addr, float value) {
  if (value >= 0.0f) atomicMax((int*)addr, __float_as_int(value));
  else               atomicMin((unsigned int*)addr, __float_as_uint(value));
}

// ---------------- stage 0: vl, goal, xin ----------------
__global__ __launch_bounds__(256)
void k_prep(const float* __restrict__ v, const float* __restrict__ labels,
            float* __restrict__ vl, float* __restrict__ goal,
            float* __restrict__ xin)
{
  int t = threadIdx.x;
  for (int idx = t; idx < N_NODES * 8; idx += blockDim.x) {
    int r = idx >> 3, c = idx & 7;
    vl[idx] = (c < CS) ? v[r * CS + c] : labels[r];
  }
  __syncthreads();
  if (t < 8) {
    float acc = 0.0f;
    for (int r = 0; r < N_NODES; ++r)
      if (labels[r] == 1.0f) acc += vl[r * 8 + t];
    goal[t] = acc;
  }
  __syncthreads();
  for (int idx = t; idx < N_NODES * 32; idx += blockDim.x) {
    int r = idx >> 5, c = idx & 31;
    float g = goal[c & 7];
    float vv = vl[r * 8 + (c & 7)];
    float d = vv - g;
    float o;
    if (c < 8)       o = vv;
    else if (c < 16) o = g;
    else if (c < 24) o = d;
    else             o = d * d;
    xin[idx] = o;
  }
}

// ---------------- hx MLP: x = mlp(xin) ----------------
__global__ __launch_bounds__(32)
void k_hx(const float* __restrict__ xin,
          const float* W0, const float* b0,
          const float* W1, const float* b1, float* __restrict__ x)
{
  __shared__ float A[16 * 32];
  __shared__ float H[16 * 64];
  int lane = threadIdx.x & 31;
  int base = blockIdx.x * 16;
  for (int idx = lane; idx < 16 * 32; idx += 32) A[idx] = xin[base * 32 + idx];
  __syncthreads();
  wave_gemm16(A, 32, W0, 64, false, 32, b0, true, H, 64, 0, 4, lane);
  __syncthreads();
  wave_gemm16(H, 64, W1, 64, false, 64, b1, false, x + (size_t)base * 64, 64, 0, 4, lane);
}

// ---------------- hy MLP: y = mlp([vj-vi, vj, vi]) ----------------
__global__ __launch_bounds__(32)
void k_hy(const float* __restrict__ vl, const int* __restrict__ e0,
          const int* __restrict__ e1,
          const float* W0, const float* b0,
          const float* W1, const float* b1, float* __restrict__ y)
{
  __shared__ float A[16 * 24];
  __shared__ float H[16 * 64];
  int lane = threadIdx.x & 31;
  int hf = lane >> 4, row = lane & 15;
  int e = blockIdx.x * 16 + row;
  int i0 = e0[e], i1 = e1[e];
  for (int cc = 0; cc < 12; ++cc) {
    int c = hf * 12 + cc;
    float o;
    if (c < 8)       o = vl[i1 * 8 + c] - vl[i0 * 8 + c];
    else if (c < 16) o = vl[i1 * 8 + (c - 8)];
    else             o = vl[i0 * 8 + (c - 16)];
    A[row * 24 + c] = o;
  }
  __syncthreads();
  wave_gemm16(A, 24, W0, 64, false, 24, b0, true, H, 64, 0, 4, lane);
  __syncthreads();
  wave_gemm16(H, 64, W1, 64, false, 64, b1, false,
              y + (size_t)blockIdx.x * 1024, 64, 0, 4, lane);
}

// ---------------- obstacle embeddings (2 blocks: node / edge) ----------------
__global__ __launch_bounds__(32)
void k_obs(const float* __restrict__ obstacles, const float* __restrict__ pos_enc,
           const float* nW0, const float* nb0, const float* nW1, const float* nb1,
           const float* eW0, const float* eb0, const float* eW1, const float* eb1,
           float* __restrict__ obs_n, float* __restrict__ obs_e)
{
  __shared__ float A[16 * 12];
  __shared__ float W0p[12 * 64];       // zero-padded first-layer weight (9 -> 12 rows)
  __shared__ float H[16 * 64];
  __shared__ float F[16 * 64];
  int lane = threadIdx.x & 31;
  bool edge = (blockIdx.x == 1);
  const float* W0 = edge ? eW0 : nW0;
  const float* B0 = edge ? eb0 : nb0;
  const float* W1 = edge ? eW1 : nW1;
  const float* B1 = edge ? eb1 : nb1;
  for (int idx = lane; idx < 16 * 12; idx += 32) {
    int r = idx / 12, c = idx % 12;
    A[idx] = (c < OBS) ? obstacles[r * OBS + c] : 0.0f;
  }
  for (int idx = lane; idx < 12 * 64; idx += 32) {
    int r = idx >> 6;
    W0p[idx] = (r < OBS) ? W0[idx] : 0.0f;
  }
  __syncthreads();
  wave_gemm16(A, 12, W0p, 64, false, 12, B0, true, H, 64, 0, 4, lane);
  __syncthreads();
  wave_gemm16(H, 64, W1, 64, false, 64, B1, false, F, 64, 0, 4, lane);
  __syncthreads();
  float* out = edge ? obs_e : obs_n;
  for (int idx = lane; idx < 16 * 64; idx += 32) out[idx] = F[idx] + pos_enc[idx];
}

// ---------------- generic 16x64 @ 64x64 (ok/ov precompute) ----------------
__global__ __launch_bounds__(32)
void k_g16(const float* __restrict__ A_g, const float* __restrict__ W,
           float* __restrict__ out)
{
  __shared__ float A[1024];
  int lane = threadIdx.x & 31;
  for (int idx = lane; idx < 1024; idx += 32) A[idx] = A_g[idx];
  __syncthreads();
  wave_gemm16(A, 64, W, 64, false, 64, nullptr, false, out, 64, 0, 4, lane);
}

// ---------------- fused attention block (in-place on data rows) ----------------
__global__ __launch_bounds__(32)
void k_att(float* __restrict__ data,
           const float* Wk, const float* Wq, const float* Wv,
           const float* ag, const float* ab,
           const float* w1W, const float* w1b,
           const float* w2W, const float* w2b,
           const float* fg, const float* fb,
           const float* __restrict__ ok, const float* __restrict__ ov)
{
  __shared__ float vc[1024], q[1024], kk[1024], vv[1024], s1[1024];
  __shared__ float att[16 * 18];
  int lane = threadIdx.x & 31;
  size_t base = (size_t)blockIdx.x * 1024;
  for (int idx = lane; idx < 1024; idx += 32) vc[idx] = data[base + idx];
  __syncthreads();
  wave_gemm16(vc, 64, Wq, 64, false, 64, nullptr, false, q,  64, 0, 4, lane);
  wave_gemm16(vc, 64, Wk, 64, false, 64, nullptr, false, kk, 64, 0, 4, lane);
  wave_gemm16(vc, 64, Wv, 64, false, 64, nullptr, false, vv, 64, 0, 4, lane);
  __syncthreads();
  if (lane < 16) {                                 // self logit: dot(vq, vk)
    float s = 0.0f;
    for (int c = 0; c < 64; ++c) s += q[lane * 64 + c] * kk[lane * 64 + c];
    att[lane * 18 + 0] = s;
  }
  __syncthreads();
  // obstacle logits: vq @ ok^T  -> att cols 1..16
  wave_gemm16(q, 64, ok, 64, true, 64, nullptr, false, att, 18, 1, 1, lane);
  __syncthreads();
  if (lane < 16) {                                 // softmax over 17 logits / TEMP
    float mx = -3.0e38f;
    for (int j = 0; j < 17; ++j) mx = fmaxf(mx, att[lane * 18 + j] * INV_TEMP);
    float s = 0.0f;
    for (int j = 0; j < 17; ++j) {
      float e = __expf(att[lane * 18 + j] * INV_TEMP - mx);
      att[lane * 18 + j] = e;
      s += e;
    }
    float inv = 1.0f / s;
    for (int j = 0; j < 17; ++j) att[lane * 18 + j] *= inv;
  }
  __syncthreads();
  for (int idx = lane; idx < 1024; idx += 32) {    // new = w0*vv + w[1:] @ ov ; +resid
    int row = idx >> 6, col = idx & 63;
    float val = att[row * 18 + 0] * vv[idx];
    for (int m = 0; m < 16; ++m) val += att[row * 18 + 1 + m] * ov[m * 64 + col];
    s1[idx] = val + vc[idx];
  }
  __syncthreads();
  if (lane < 16) {                                 // LayerNorm -> vc
    int row = lane;
    float m = 0.0f;
    for (int c = 0; c < 64; ++c) m += s1[row * 64 + c];
    m *= (1.0f / 64.0f);
    float var = 0.0f;
    for (int c = 0; c < 64; ++c) { float d = s1[row * 64 + c] - m; var += d * d; }
    var *= (1.0f / 64.0f);
    float inv = 1.0f / sqrtf(var + LN_EPS);
    for (int c = 0; c < 64; ++c)
      vc[row * 64 + c] = (s1[row * 64 + c] - m) * inv * ag[c] + ab[c];
  }
  __syncthreads();
  wave_gemm16(vc, 64, w1W, 64, false, 64, w1b, true, q, 64, 0, 4, lane);   // h
  __syncthreads();
  wave_gemm16(q, 64, w2W, 64, false, 64, w2b, false, s1, 64, 0, 4, lane);  // f
  __syncthreads();
  if (lane < 16) {                                 // LN(f + vc) -> out
    int row = lane;
    float m = 0.0f;
    for (int c = 0; c < 64; ++c) { s1[row * 64 + c] += vc[row * 64 + c]; m += s1[row * 64 + c]; }
    m *= (1.0f / 64.0f);
    float var = 0.0f;
    for (int c = 0; c < 64; ++c) { float d = s1[row * 64 + c] - m; var += d * d; }
    var *= (1.0f / 64.0f);
    float inv = 1.0f / sqrtf(var + LN_EPS);
    for (int c = 0; c < 64; ++c)
      data[base + row * 64 + c] = (s1[row * 64 + c] - m) * inv * fg[c] + fb[c];
  }
}

// ---------------- fx message MLP + segment-max (atomic) ----------------
__global__ __launch_bounds__(32)
void k_msg(const float* __restrict__ x, const float* __restrict__ y,
           const int* __restrict__ e0, const int* __restrict__ e1,
           const float* W0, const float* b0,
           const float* W1, const float* b1, float* __restrict__ agg)
{
  __shared__ float A[16 * 256];
  __shared__ float H[1024];
  __shared__ float F[1024];
  __shared__ int tgt[16];
  int lane = threadIdx.x & 31;
  int hf = lane >> 4, row = lane & 15;
  int e = blockIdx.x * 16 + row;
  int isrc = e0[e], itgt = e1[e];
  if (hf == 0) tgt[row] = itgt;
  for (int cc = 0; cc < 32; ++cc) {
    int c = hf * 32 + cc;
    float xj = x[isrc * 64 + c];      // x_j = x[e0]
    float xi = x[itgt * 64 + c];      // x_i = x[e1]
    A[row * 256 + c]       = xj - xi;
    A[row * 256 + 64 + c]  = xj;
    A[row * 256 + 128 + c] = xi;
    A[row * 256 + 192 + c] = y[(size_t)e * 64 + c];
  }
  __syncthreads();
  wave_gemm16(A, 256, W0, 64, false, 256, b0, true, H, 64, 0, 4, lane);
  __syncthreads();
  wave_gemm16(H, 64, W1, 64, false, 64, b1, false, F, 64, 0, 4, lane);
  __syncthreads();
  for (int idx = lane; idx < 1024; idx += 32) {
    int r = idx >> 6, c = idx & 63;
    atomicMaxF(&agg[tgt[r] * 64 + c], F[idx]);
  }
}

// ---------------- fy edge update: y = max(y, mlp) ----------------
__global__ __launch_bounds__(32)
void k_yupd(const float* __restrict__ x, float* __restrict__ y,
            const int* __restrict__ e0, const int* __restrict__ e1,
            const float* W0, const float* b0,
            const float* W1, const float* b1)
{
  __shared__ float A[16 * 192];
  __shared__ float H[1024];
  __shared__ float F[1024];
  int lane = threadIdx.x & 31;
  int hf = lane >> 4, row = lane & 15;
  int e = blockIdx.x * 16 + row;
  int i0 = e0[e], i1 = e1[e];
  for (int cc = 0; cc < 32; ++cc) {
    int c = hf * 32 + cc;
    float xj = x[i1 * 64 + c];        // xj2 = x[e1]
    float xi = x[i0 * 64 + c];        // xi2 = x[e0]
    A[row * 192 + c]       = xj - xi;
    A[row * 192 + 64 + c]  = xj;
    A[row * 192 + 128 + c] = xi;
  }
  __syncthreads();
  wave_gemm16(A, 192, W0, 64, false, 192, b0, true, H, 64, 0, 4, lane);
  __syncthreads();
  wave_gemm16(H, 64, W1, 64, false, 64, b1, false, F, 64, 0, 4, lane);
  __syncthreads();
  for (int idx = lane; idx < 1024; idx += 32) {
    size_t o = (size_t)blockIdx.x * 1024 + idx;
    y[o] = fmaxf(y[o], F[idx]);
  }
}

// ---------------- elementwise helpers ----------------
__global__ __launch_bounds__(256)
void k_fill(float* __restrict__ p, long n, float val) {
  long i = (long)blockIdx.x * blockDim.x + threadIdx.x;
  long stride = (long)gridDim.x * blockDim.x;
  for (; i < n; i += stride) p[i] = val;
}
__global__ __launch_bounds__(256)
void k_zero4(float4* __restrict__ p, long n4) {
  long i = (long)blockIdx.x * blockDim.x + threadIdx.x;
  long stride = (long)gridDim.x * blockDim.x;
  float4 z; z.x = z.y = z.z = z.w = 0.0f;
  for (; i < n4; i += stride) p[i] = z;
}
__global__ __launch_bounds__(256)
void k_xupd(float* __restrict__ x, const float* __restrict__ agg, int n) {
  int i = blockIdx.x * blockDim.x + threadIdx.x;
  if (i >= n) return;
  float a = agg[i];
  unsigned u = __float_as_uint(a);
  if ((u & 0x7f800000u) == 0x7f800000u) a = 0.0f;   // non-finite -> 0
  x[i] = fmaxf(x[i], a);
}
__global__ __launch_bounds__(256)
void k_scatter(const float* __restrict__ y, const int* __restrict__ e0,
               const int* __restrict__ e1, float* __restrict__ out) {
  long i = (long)blockIdx.x * blockDim.x + threadIdx.x;
  long stride = (long)gridDim.x * blockDim.x;
  for (; i < (long)NE * 64; i += stride) {
    long e = i >> 6; int c = (int)(i & 63);
    out[((size_t)e0[e] * N_NODES + (size_t)e1[e]) * 64 + c] = y[i];
  }
}
__global__ __launch_bounds__(256)
void k_copy(const float* __restrict__ src, float* __restrict__ dst, int n) {
  int i = blockIdx.x * blockDim.x + threadIdx.x;
  if (i < n) dst[i] = src[i];
}

// --------------------------------------------------------------------
// d_in layout (depth-first insertion-order flattening of setup_inputs()):
//  0 v, 1 labels, 2 obstacles, 3 pos_enc, 4 edge_index(int32 [2,NE]), 5 loop,
//  6.. params: hx{W0,b0,W1,b1} hy{..} fx{..} fy{..} obs_node{..} obs_edge{..}
//  then node_att[3] blocks (11 leaves each: Wk,Wq,Wv,att.g,att.b,
//  w1.W,w1.b,w2.W,w2.b,ff.g,ff.b), then edge_att[3] likewise.
// --------------------------------------------------------------------
extern "C" void kernel_launch(void* const* d_in, const int* in_sizes, int n_in,
                              void* d_out, int out_size, void* d_ws, size_t ws_size,
                              hipStream_t stream) {
  (void)in_sizes; (void)n_in; (void)out_size; (void)ws_size;
  auto F = [&](int i) { return (const float*)d_in[i]; };
  const int* ei = (const int*)d_in[4];
  const int* e0 = ei;
  const int* e1 = ei + NE;

  const int HX = 6, HY = 10, FX = 14, FY = 18, ON = 22, OE = 26;
  const int NB = 30, EB = 63;      // node_att / edge_att block bases, stride 11

  // workspace partition (floats)
  float* wf    = (float*)d_ws;
  float* vl    = wf;                      // 1200*8
  float* goal  = vl + N_NODES * 8;        // 8
  float* xin   = goal + 8;                // 1200*32
  float* x     = xin + N_NODES * 32;      // 1200*64
  float* y     = x + N_NODES * 64;        // 38400*64
  float* obs_n = y + (size_t)NE * 64;     // 16*64
  float* obs_e = obs_n + 1024;            // 16*64
  float* okov  = obs_e + 1024;            // 12 * 16*64
  float* agg   = okov + 12 * 1024;        // 1200*64

  float* fout = (float*)d_out;
  const long edge_elems = (long)N_NODES * N_NODES * 64;

  // zero the dense [N,N,64] output region (independent of the pipeline)
  k_zero4<<<8192, 256, 0, stream>>>((float4*)fout, edge_elems / 4);

  // node features and initial embeddings
  k_prep<<<1, 256, 0, stream>>>(F(0), F(1), vl, goal, xin);
  k_hx<<<N_NODES / 16, 32, 0, stream>>>(xin, F(HX), F(HX + 1), F(HX + 2), F(HX + 3), x);
  k_hy<<<NE / 16, 32, 0, stream>>>(vl, e0, e1, F(HY), F(HY + 1), F(HY + 2), F(HY + 3), y);
  k_obs<<<2, 32, 0, stream>>>(F(2), F(3),
                              F(ON), F(ON + 1), F(ON + 2), F(ON + 3),
                              F(OE), F(OE + 1), F(OE + 2), F(OE + 3),
                              obs_n, obs_e);

  // precompute ok/ov for each attention block (k_g16: obs @ W)
  for (int i = 0; i < 3; ++i) {
    k_g16<<<1, 32, 0, stream>>>(obs_n, F(NB + 11 * i + 0), okov + (2 * i) * 1024);      // node ok
    k_g16<<<1, 32, 0, stream>>>(obs_n, F(NB + 11 * i + 2), okov + (2 * i + 1) * 1024);  // node ov
    k_g16<<<1, 32, 0, stream>>>(obs_e, F(EB + 11 * i + 0), okov + (6 + 2 * i) * 1024);  // edge ok
    k_g16<<<1, 32, 0, stream>>>(obs_e, F(EB + 11 * i + 2), okov + (7 + 2 * i) * 1024);  // edge ov
  }

  // 3 attention blocks on x (node) and y (edge)
  for (int i = 0; i < 3; ++i) {
    int nb = NB + 11 * i, eb = EB + 11 * i;
    k_att<<<N_NODES / 16, 32, 0, stream>>>(x,
        F(nb + 0), F(nb + 1), F(nb + 2), F(nb + 3), F(nb + 4),
        F(nb + 5), F(nb + 6), F(nb + 7), F(nb + 8), F(nb + 9), F(nb + 10),
        okov + (2 * i) * 1024, okov + (2 * i + 1) * 1024);
    k_att<<<NE / 16, 32, 0, stream>>>(y,
        F(eb + 0), F(eb + 1), F(eb + 2), F(eb + 3), F(eb + 4),
        F(eb + 5), F(eb + 6), F(eb + 7), F(eb + 8), F(eb + 9), F(eb + 10),
        okov + (6 + 2 * i) * 1024, okov + (7 + 2 * i) * 1024);
  }

  // message-passing loop (loop == 3 in setup_inputs, compile-time constant)
  for (int it = 0; it < LOOPN; ++it) {
    k_fill<<<300, 256, 0, stream>>>(agg, (long)N_NODES * 64, -__builtin_huge_valf());
    k_msg<<<NE / 16, 32, 0, stream>>>(x, y, e0, e1,
                                      F(FX), F(FX + 1), F(FX + 2), F(FX + 3), agg);
    k_xupd<<<(N_NODES * 64 + 255) / 256, 256, 0, stream>>>(x, agg, N_NODES * 64);
    k_yupd<<<NE / 16, 32, 0, stream>>>(x, y, e0, e1,
                                       F(FY), F(FY + 1), F(FY + 2), F(FY + 3));
  }

  // outputs: dense edge_feat scatter, then x appended
  k_scatter<<<9600, 256, 0, stream>>>(y, e0, e1, fout);
  k_copy<<<(N_NODES * 64 + 255) / 256, 256, 0, stream>>>(x, fout + edge_elems, N_NODES * 64);
}